// BoosterModelLoss_55396488184123
// MI455X (gfx1250) — compile-verified
//
#include <hip/hip_runtime.h>
#include <math.h>

// ---- problem constants (from setup_inputs) ----
#define BATCH   32
#define TDIM    512
#define VDIM    2048
#define SGL     32      // glosses per batch
#define LWORD   64
#define WVOC    4096
#define NTAGS   12
#define LEXT    65      // 2*S+1
#define NEGV    (-100000.0f)

typedef __attribute__((ext_vector_type(2))) float v2f;
typedef __attribute__((ext_vector_type(8))) float v8f;

__device__ __forceinline__ float lae(float a, float b) {
    float M = fmaxf(a, b);
    return M + log1pf(__expf(-fabsf(a - b)));
}

__global__ void zero_acc_kernel(float* acc) {
    acc[0] = 0.0f;  // recognition loss accumulator
    acc[1] = 0.0f;  // translation loss accumulator
}

// -----------------------------------------------------------------------------
// Phase 1: per 16-row tile of (B*T, V):
//   - WMMA fp32 GEMM against one-hot(tag) (+ ones column 12) -> per-tag sums
//   - means, boosted logsumexp (online, wave-level), emit gather (B,T,65)
// -----------------------------------------------------------------------------
__global__ __launch_bounds__(256) void booster_emit_kernel(
    const float* __restrict__ scores,   // (B*T, V)
    const int*   __restrict__ tag_ids,  // (V,)
    const int*   __restrict__ glosses,  // (B, S)
    float*       __restrict__ emit)     // (B*T, 65)
{
    __shared__ float stage[16][132];       // 16 x 128 chunk, padded stride
    __shared__ int   tags[VDIM];
    __shared__ float wavebins[8][256];
    __shared__ float bins[16][16];
    __shared__ float fmeans[16][NTAGS];    // 0.1 * means
    __shared__ int   counts[NTAGS];

    const int tid  = threadIdx.x;
    const int lane = tid & 31;
    const int wave = tid >> 5;
    const int row_base = blockIdx.x * 16;

    if (tid < NTAGS) counts[tid] = 0;
    __syncthreads();
    for (int i = tid; i < VDIM; i += 256) {
        int tg = tag_ids[i];
        tags[i] = tg;
        atomicAdd(&counts[tg], 1);
    }
    __syncthreads();

    // ---- WMMA bin sums: C[m][n] = sum_v scores[row_base+m][v] * B(v,n) ----
    // B(v,n) = (tag[v]==n) for n<12 ; B(v,12)=1 (row-total column)
    v8f c = {};
    const int m_a = lane & 15;             // A row handled by this lane
    const int ko  = (lane >> 4) << 1;      // K sub-offset (0 or 2)
    const int n_b = lane & 15;             // B column handled by this lane
    for (int oc = 0; oc < VDIM / 128; ++oc) {
        { // cooperative, coalesced load of the 16x128 chunk
            int r   = tid >> 4;
            int seg = tid & 15;
            const float* g = scores + (size_t)(row_base + r) * VDIM + oc * 128 + seg * 8;
#pragma unroll
            for (int j = 0; j < 8; ++j) stage[r][seg * 8 + j] = g[j];
        }
        __syncthreads();
#pragma unroll
        for (int s = 0; s < 4; ++s) {
            int kc = (wave * 4 + s) * 4;   // each wave owns 4 K-chunks of 4
            v2f a;
            a[0] = stage[m_a][kc + ko];
            a[1] = stage[m_a][kc + ko + 1];
            int kg = oc * 128 + kc + ko;
            int t0 = tags[kg];
            int t1 = tags[kg + 1];
            v2f bf;
            bf[0] = (t0 == n_b || n_b == 12) ? 1.0f : 0.0f;
            bf[1] = (t1 == n_b || n_b == 12) ? 1.0f : 0.0f;
            // D = A(16x4, f32) * B(4x16, f32) + C  -> v_wmma_f32_16x16x4_f32
            c = __builtin_amdgcn_wmma_f32_16x16x4_f32(
                    false, a, false, bf, (short)0, c, false, false);
        }
        __syncthreads();
    }

    // ---- reduce partial C across the 8 waves ----
#pragma unroll
    for (int r = 0; r < 8; ++r) wavebins[wave][r * 32 + lane] = c[r];
    __syncthreads();
    {
        float s = 0.0f;
#pragma unroll
        for (int w = 0; w < 8; ++w) s += wavebins[w][tid];
        int r = tid >> 5, ln = tid & 31;
        int M = (ln < 16) ? r : r + 8;     // C layout: VGPR r, lanes 0-15 -> M=r, 16-31 -> M=8+r
        int N = ln & 15;
        bins[M][N] = s;
    }
    __syncthreads();
    if (tid < 16 * NTAGS) {
        int m = tid / NTAGS, k = tid % NTAGS;
        float mv = (k == NTAGS - 1)
                     ? bins[m][12] * (1.0f / (float)VDIM)                 // overall row mean
                     : bins[m][k] / fmaxf((float)counts[k], 1.0f);
        fmeans[m][k] = 0.1f * mv;                                         // FACTOR folded in
    }
    __syncthreads();

    // ---- per-row logsumexp of boosted scores + emit gather ----
#pragma unroll
    for (int rr = 0; rr < 2; ++rr) {
        int m = wave * 2 + rr;                  // wave handles rows 2w, 2w+1
        size_t row = (size_t)(row_base + m);
        const float* g = scores + row * VDIM;
        float mx = -INFINITY, sm = 0.0f;        // online logsumexp
        for (int it = 0; it < VDIM / 32; ++it) {
            int v = it * 32 + lane;
            float x = g[v] + fmeans[m][tags[v]];
            float M2 = fmaxf(mx, x);
            sm = sm * __expf(mx - M2) + __expf(x - M2);
            mx = M2;
        }
#pragma unroll
        for (int off = 16; off > 0; off >>= 1) {
            float mo = __shfl_xor(mx, off, 32);
            float so = __shfl_xor(sm, off, 32);
            float M2 = fmaxf(mx, mo);
            sm = sm * __expf(mx - M2) + so * __expf(mo - M2);
            mx = M2;
        }
        float logZ = mx + __logf(sm);
        int b = (int)(row >> 9);                // row / TDIM
        for (int l = lane; l < LEXT; l += 32) {
            int ext = (l & 1) ? glosses[b * SGL + ((l - 1) >> 1)] : 0;
            emit[row * LEXT + l] = g[ext] + fmeans[m][tags[ext]] - logZ;
        }
    }
}

// -----------------------------------------------------------------------------
// Phase 2: CTC forward recursion, one block per batch element
// -----------------------------------------------------------------------------
__global__ __launch_bounds__(128) void ctc_kernel(
    const float* __restrict__ emit, const int* __restrict__ glosses,
    const int* __restrict__ frames_len, const int* __restrict__ glosses_len,
    float* __restrict__ acc_rec)
{
    __shared__ int           ext_s[LEXT];
    __shared__ unsigned char skip_s[LEXT];
    __shared__ float         alpha[LEXT];
    const int tid = threadIdx.x;
    const int b   = blockIdx.x;

    if (tid < LEXT)
        ext_s[tid] = (tid & 1) ? glosses[b * SGL + ((tid - 1) >> 1)] : 0;
    __syncthreads();
    if (tid < LEXT) {
        skip_s[tid] = (tid >= 2) && (tid & 1) && (ext_s[tid] != ext_s[tid - 2]);
        alpha[tid]  = (tid < 2) ? emit[((size_t)b * TDIM) * LEXT + tid] : NEGV;
    }
    __syncthreads();
    const int flen = frames_len[b];
    for (int t = 1; t < TDIM; ++t) {
        float nv = 0.0f;
        if (tid < LEXT) {
            float a0 = alpha[tid];
            float a1 = (tid >= 1) ? alpha[tid - 1] : NEGV;
            float a2 = (tid >= 2 && skip_s[tid]) ? alpha[tid - 2] : NEGV;
            float v  = lae(lae(a0, a1), a2) + emit[((size_t)b * TDIM + t) * LEXT + tid];
            nv = (t < flen) ? v : a0;
        }
        __syncthreads();
        if (tid < LEXT) alpha[tid] = nv;
        __syncthreads();
    }
    if (tid == 0) {
        int tl = glosses_len[b];
        float loss = -lae(alpha[2 * tl], alpha[2 * tl - 1]);
        if (!isfinite(loss)) loss = 0.0f;
        atomicAdd(acc_rec, loss / (float)tl * (1.0f / (float)BATCH));
    }
}

// -----------------------------------------------------------------------------
// Phase 3: NLL gather (only 32*63 values of words_output are ever touched)
// -----------------------------------------------------------------------------
__global__ void nll_kernel(const float* __restrict__ lp, const int* __restrict__ words,
                           float* __restrict__ acc_trans)
{
    int b = blockIdx.x, j = threadIdx.x;
    if (j < LWORD - 1) {
        int tgt = words[b * LWORD + j + 1];
        if (tgt != 1) {  // IGNORE_INDEX
            float p = lp[((size_t)b * LWORD + j) * WVOC + tgt];
            atomicAdd(acc_trans, -p * (1.0f / (float)BATCH));
        }
    }
}

__global__ void finalize_kernel(const float* __restrict__ acc, float* __restrict__ out) {
    float rec = acc[0], trans = acc[1];
    out[0] = rec + trans;   // total
    out[1] = rec;           // recognition_loss
    out[2] = trans;         // translation_loss
}

extern "C" void kernel_launch(void* const* d_in, const int* in_sizes, int n_in,
                              void* d_out, int out_size, void* d_ws, size_t ws_size,
                              hipStream_t stream) {
    const float* scores       = (const float*)d_in[0];  // (B,T,V)
    const float* words_output = (const float*)d_in[1];  // (B,L,W) log-probs
    const int*   glosses      = (const int*)d_in[2];    // (B,S)
    const int*   words        = (const int*)d_in[3];    // (B,L)
    const int*   frames_len   = (const int*)d_in[4];    // (B,)
    const int*   glosses_len  = (const int*)d_in[5];    // (B,)
    const int*   tag_ids      = (const int*)d_in[6];    // (V,)
    float* out  = (float*)d_out;
    float* acc  = (float*)d_ws;            // [0]=rec, [1]=trans
    float* emit = (float*)d_ws + 64;       // 256B offset; (B*T*65) floats ~ 4.3 MB

    zero_acc_kernel<<<1, 1, 0, stream>>>(acc);
    booster_emit_kernel<<<(BATCH * TDIM) / 16, 256, 0, stream>>>(scores, tag_ids, glosses, emit);
    ctc_kernel<<<BATCH, 128, 0, stream>>>(emit, glosses, frames_len, glosses_len, acc);
    nll_kernel<<<BATCH, LWORD, 0, stream>>>(words_output, words, acc + 1);
    finalize_kernel<<<1, 1, 0, stream>>>(acc, out);
}